// w_angular_j1_v1_41523743817758
// MI455X (gfx1250) — compile-verified
//
#include <hip/hip_runtime.h>
#include <hip/hip_bf16.h>
#include <math.h>

typedef __attribute__((ext_vector_type(2))) float v2f;
typedef __attribute__((ext_vector_type(8))) float v8f;

#define C_CLASSES 1000
#define C_PAD     1024
#define TILE      16
#define NT        (C_PAD / TILE)   /* 64 tiles per dim, 4096 total */
#define SORT_CAP  2048

__device__ __forceinline__ unsigned int enc_f32(float f) {
  unsigned int u = __float_as_uint(f);
  return (u & 0x80000000u) ? ~u : (u | 0x80000000u);
}
__device__ __forceinline__ float dec_f32(unsigned int e) {
  unsigned int u = (e & 0x80000000u) ? (e ^ 0x80000000u) : ~e;
  return __uint_as_float(u);
}

// ---------------- init ----------------
__global__ void k_init(int* counts, int* cursor) {
  int t = threadIdx.x;
  counts[t] = 0;
  cursor[t] = 0;
}

// ---------------- label histogram ----------------
__global__ void k_hist(const int* __restrict__ labels, int* counts, int n) {
  int i = blockIdx.x * blockDim.x + threadIdx.x;
  if (i < n) atomicAdd(&counts[labels[i]], 1);
}

// ---------------- exclusive scan over 1024 class counts ----------------
__global__ void k_scan(const int* __restrict__ counts, int* offsets) {
  __shared__ int s[C_PAD];
  int t = threadIdx.x;
  s[t] = counts[t];
  __syncthreads();
  for (int off = 1; off < C_PAD; off <<= 1) {
    int v = (t >= off) ? s[t - off] : 0;
    __syncthreads();
    s[t] += v;
    __syncthreads();
  }
  offsets[t] = s[t] - counts[t];   // exclusive prefix
}

// ---------------- scatter row ids into class buckets ----------------
__global__ void k_scatter(const int* __restrict__ labels, const int* __restrict__ offsets,
                          int* cursor, int* rowidx, int n) {
  int i = blockIdx.x * blockDim.x + threadIdx.x;
  if (i < n) {
    int c = labels[i];
    int pos = atomicAdd(&cursor[c], 1);
    rowidx[offsets[c] + pos] = i;
  }
}

// ---------------- per-class deterministic sum (one block per class) ----------------
__global__ void k_classsum(const float* __restrict__ feats, const int* __restrict__ rowidx,
                           const int* __restrict__ counts, const int* __restrict__ offsets,
                           float* sums, int D) {
  __shared__ int idx[SORT_CAP];
  __shared__ int srt[SORT_CAP];
  int c = blockIdx.x, t = threadIdx.x;
  int n = counts[c], base = offsets[c];
  float acc = 0.f;
  if (n <= SORT_CAP) {
    for (int e = t; e < n; e += blockDim.x) idx[e] = rowidx[base + e];
    __syncthreads();
    // rank-sort: fixed summation order regardless of scatter arrival order
    for (int e = t; e < n; e += blockDim.x) {
      int v = idx[e], r = 0;
      for (int f = 0; f < n; ++f) r += (idx[f] < v);
      srt[r] = v;
    }
    __syncthreads();
    for (int r = 0; r < n; ++r)
      acc += feats[(size_t)srt[r] * D + t];      // coalesced 1KB row reads
  } else {
    for (int r = 0; r < n; ++r)
      acc += feats[(size_t)rowidx[base + r] * D + t];
  }
  sums[(size_t)c * D + t] = acc;
}

// ---------------- global mean ----------------
__global__ void k_gmean(const float* __restrict__ sums, float* gmean, int D, float invN) {
  int d = threadIdx.x;
  float s = 0.f;
  for (int c = 0; c < C_CLASSES; ++c) s += sums[(size_t)c * D + d];
  gmean[d] = s * invN;
}

// ---------------- normalize centroids + sb list (one block per padded class) ----------------
__global__ void k_norm(const float* __restrict__ sums, const int* __restrict__ counts,
                       const float* __restrict__ gmean, float* mn, float* sb,
                       int D, float invN) {
  __shared__ float red[256];
  int c = blockIdx.x, t = threadIdx.x;
  float m = 0.f; int cnt = 0;
  if (c < C_CLASSES) {
    cnt = counts[c];
    m = sums[(size_t)c * D + t] / fmaxf((float)cnt, 1.0f);
  }
  red[t] = m * m;
  __syncthreads();
  for (int s = 128; s > 0; s >>= 1) { if (t < s) red[t] += red[t + s]; __syncthreads(); }
  float nrm2 = red[0];
  __syncthreads();
  float dv = (c < C_CLASSES) ? (m - gmean[t]) : 0.f;
  red[t] = dv * dv;
  __syncthreads();
  for (int s = 128; s > 0; s >>= 1) { if (t < s) red[t] += red[t + s]; __syncthreads(); }
  float db2 = red[0];
  mn[(size_t)c * D + t] = (c < C_CLASSES) ? (m / sqrtf(nrm2)) : 0.f;
  if (t == 0) sb[c] = (c < C_CLASSES) ? ((float)cnt * invN * db2) : 0.f;
}

// ---------------- fp32 WMMA Gram + per-tile argmin (one wave per 16x16 tile) ----------------
// Upper triangle only: dist is symmetric and the reference's first-row-major argmin
// always lands at (i<j); masking gi<gj preserves exact tie-break (min key wins).
__global__ void k_gram_argmin(const float* __restrict__ mn, unsigned long long* tilemin, int D) {
  int lane = threadIdx.x;                 // wave32, EXEC all ones for WMMA
  int tile = blockIdx.x;
  int tr = tile >> 6, tc = tile & (NT - 1);
  if (tr > tc) {                          // lower triangle: nothing to do
    if (lane == 0) tilemin[tile] = ~0ull;
    return;
  }
  int rowBase = tr * TILE, colBase = tc * TILE;
  int m15 = lane & 15;
  int ko  = (lane >> 4) * 2;              // lanes 16-31 carry K=2,3 (ISA 7.12.2 f32 A layout)
  const float* arow = mn + (size_t)(rowBase + m15) * D + ko;   // A: 16x4 tile of Mn
  const float* brow = mn + (size_t)(colBase + m15) * D + ko;   // B: 4x16 tile of Mn^T
  v8f acc = {};
  #pragma unroll 8
  for (int k = 0; k < D; k += 4) {
    v2f a = *(const v2f*)(arow + k);
    v2f b = *(const v2f*)(brow + k);
    acc = __builtin_amdgcn_wmma_f32_16x16x4_f32(
        /*neg_a=*/false, a, /*neg_b=*/false, b,
        /*c_mod=*/(short)0, acc, /*reuse_a=*/false, /*reuse_b=*/false);
  }
  // C/D layout: VGPR v, lanes 0-15 -> M=v ; lanes 16-31 -> M=8+v ; N = lane&15
  int moff = (lane >> 4) * 8;
  int gj = colBase + m15;
  unsigned long long best = ~0ull;
  #pragma unroll
  for (int v = 0; v < 8; ++v) {
    int gi = rowBase + moff + v;
    float g = fminf(fmaxf(acc[v], -1.0f), 1.0f);
    float dist = 1.0f - g;
    if (gi < gj && gj < C_CLASSES) {
      unsigned long long kk = ((unsigned long long)enc_f32(dist) << 32)
                            | (unsigned int)(gi * C_CLASSES + gj);
      best = (kk < best) ? kk : best;
    }
  }
  // wave-level 64-bit min reduction -> single store per tile (no global atomics)
  #pragma unroll
  for (int off = 16; off > 0; off >>= 1) {
    unsigned long long other = __shfl_down(best, off, 32);
    best = (other < best) ? other : best;
  }
  if (lane == 0) tilemin[tile] = best;
}

// ---------------- contention-free final reduction + scalar loss ----------------
__global__ void k_final(const unsigned long long* __restrict__ tilemin,
                        const float* __restrict__ sb, float* out) {
  __shared__ unsigned long long rk[1024];
  __shared__ float rs[1024];
  int t = threadIdx.x;
  unsigned long long b = ~0ull;
  for (int i = t; i < NT * NT; i += 1024) {
    unsigned long long v = tilemin[i];
    b = (v < b) ? v : b;
  }
  rk[t] = b;
  rs[t] = sb[t];                           // sb[1000..1023] written as 0 by k_norm
  __syncthreads();
  for (int s = 512; s > 0; s >>= 1) {
    if (t < s) {
      if (rk[t + s] < rk[t]) rk[t] = rk[t + s];
      rs[t] += rs[t + s];
    }
    __syncthreads();
  }
  if (t == 0) {
    unsigned long long k = rk[0];
    float md = (k == ~0ull) ? rs[0] : dec_f32((unsigned int)(k >> 32));
    out[0] = logf(1.0f / (md + 1e-6f) + 1.0f);
  }
}

extern "C" void kernel_launch(void* const* d_in, const int* in_sizes, int n_in,
                              void* d_out, int out_size, void* d_ws, size_t ws_size,
                              hipStream_t stream) {
  (void)n_in; (void)out_size; (void)ws_size;
  const float* feats  = (const float*)d_in[0];
  const int*   labels = (const int*)d_in[1];
  // d_in[2] (conf_mat) is unused by the reference.

  int N = in_sizes[1];
  int D = in_sizes[0] / N;                 // 256
  float invN = 1.0f / (float)N;

  size_t o = 0;
  auto take = [&](size_t nbytes) -> char* {
    char* p = (char*)d_ws + o;
    o += (nbytes + 255) & ~(size_t)255;
    return p;
  };
  unsigned long long* tilemin = (unsigned long long*)take(8 * (size_t)NT * NT);
  int*   counts  = (int*)take(4 * (size_t)C_PAD);
  int*   offsets = (int*)take(4 * (size_t)C_PAD);
  int*   cursor  = (int*)take(4 * (size_t)C_PAD);
  float* gmean   = (float*)take(4 * (size_t)D);
  float* sb      = (float*)take(4 * (size_t)C_PAD);
  float* sums    = (float*)take(4 * (size_t)C_PAD * D);
  float* mnrm    = (float*)take(4 * (size_t)C_PAD * D);
  int*   rowidx  = (int*)take(4 * (size_t)N);

  int nb = (N + 255) / 256;
  k_init    <<<1, C_PAD, 0, stream>>>(counts, cursor);
  k_hist    <<<nb, 256, 0, stream>>>(labels, counts, N);
  k_scan    <<<1, C_PAD, 0, stream>>>(counts, offsets);
  k_scatter <<<nb, 256, 0, stream>>>(labels, offsets, cursor, rowidx, N);
  k_classsum<<<C_CLASSES, D, 0, stream>>>(feats, rowidx, counts, offsets, sums, D);
  k_gmean   <<<1, D, 0, stream>>>(sums, gmean, D, invN);
  k_norm    <<<C_PAD, D, 0, stream>>>(sums, counts, gmean, mnrm, sb, D, invN);
  k_gram_argmin<<<NT * NT, 32, 0, stream>>>(mnrm, tilemin, D);
  k_final   <<<1, 1024, 0, stream>>>(tilemin, sb, (float*)d_out);
}